// KerasGNNLayerWrapper_66159676227859
// MI455X (gfx1250) — compile-verified
//
#include <hip/hip_runtime.h>

// ---------------------------------------------------------------------------
// GNN layer for MI455X (gfx1250): gather -> Dense(144x96)+relu+LN ->
// segment-sum -> Dense(160x96)+relu+LN.  GEMMs use V_WMMA_F32_16X16X4_F32
// with 32-row M tiles per wave and software-pipelined LDS fragment loads.
// ---------------------------------------------------------------------------

#define BB      8
#define NN      5000
#define EE      40000
#define FF      64
#define FEF     16
#define FIL     96
#define KM      144          // message GEMM K  (2F + FE)
#define KU      160          // update GEMM K   (F + FILTERS)
#define AM_STR  148          // padded LDS row stride (floats), msg kernel (16B aligned, conflict-free)
#define AU_STR  164          // padded LDS row stride (floats), upd kernel
#define ETP32   1250         // edge tiles (of 32) per batch
#define NTP32   157          // node tiles (of 32) per batch (ceil(5000/32))
#define LN_EPS  1e-3f

typedef float v8f __attribute__((ext_vector_type(8)));
typedef float v2f __attribute__((ext_vector_type(2)));

__global__ void __launch_bounds__(256) zero_f32(float* __restrict__ p, int n) {
    int i = blockIdx.x * 256 + threadIdx.x;
    if (i < n) p[i] = 0.0f;
}

#define WMMA_F32(A, Bf, C) \
    __builtin_amdgcn_wmma_f32_16x16x4_f32(false, (A), false, (Bf), (short)0, (C), false, false)

// ---------------------------------------------------------------------------
// Message kernel: 8 waves/block, one 32-edge tile per wave. grid = 1250.
// ---------------------------------------------------------------------------
__global__ void __launch_bounds__(256) msg_kernel(
    const float* __restrict__ nodes,      // [B][N][64]
    const float* __restrict__ efeat,      // [B][E][16]
    const int*   __restrict__ edges,      // [B][E][2]
    const float* __restrict__ Wm,         // [144][96] row-major
    const float* __restrict__ bm,         // [96]
    const float* __restrict__ gamma,      // [96]
    const float* __restrict__ beta,       // [96]
    float*       __restrict__ msg_out,    // [B][E][96]
    float*       __restrict__ agg)        // [B][N][96]
{
    extern __shared__ float smem[];
    float* Wt    = smem;                              // 96 * 148 floats  (W^T)
    float* Abuf  = Wt + FIL * AM_STR;                 // 256 * 148 floats
    int*   eidx  = (int*)(Abuf + 256 * AM_STR);       // 512 ints (src,dst)*256
    float* stats = (float*)(eidx + 512);              // 256 * 2 floats

    const int tid  = threadIdx.x;
    const int wave = tid >> 5;
    const int lane = tid & 31;
    const int ln16 = lane & 15;
    const int hi   = lane >> 4;

    const int t  = blockIdx.x * 8 + wave;
    const int b  = t / ETP32;
    const int e0 = (t % ETP32) * 32;

    // ---- stage edge endpoint pairs (one int2 per thread = 256 edges) -------
    {
        int r  = tid;                                 // block row 0..255
        int tw = blockIdx.x * 8 + (r >> 5);
        int bb = tw / ETP32;
        int ee = (tw % ETP32) * 32 + (r & 31);
        ((int2*)eidx)[r] = ((const int2*)edges)[(size_t)bb * EE + ee];
    }
    __syncthreads();

    // ---- stage W^T (Wt[n][k]) ----------------------------------------------
    for (int i = tid; i < KM * FIL; i += 256) {
        int k = i / FIL, n = i % FIL;
        Wt[n * AM_STR + k] = Wm[i];
    }

    // ---- gather A rows (one 144-float row per thread, float4 copies) -------
    {
        int r  = tid;                                 // 0..255
        int tw = blockIdx.x * 8 + (r >> 5);
        int bb = tw / ETP32;
        int ee = (tw % ETP32) * 32 + (r & 31);
        int src = eidx[2 * r + 0];
        int dst = eidx[2 * r + 1];
        const float4* s4 = (const float4*)(nodes + ((size_t)bb * NN + src) * FF);
        const float4* d4 = (const float4*)(nodes + ((size_t)bb * NN + dst) * FF);
        const float4* e4 = (const float4*)(efeat + ((size_t)bb * EE + ee) * FEF);
        float4* a4 = (float4*)(Abuf + (size_t)(r >> 5) * 32 * AM_STR + (r & 31) * AM_STR);
        #pragma unroll
        for (int i = 0; i < 16; ++i) a4[i]      = s4[i];
        #pragma unroll
        for (int i = 0; i < 16; ++i) a4[16 + i] = d4[i];
        #pragma unroll
        for (int i = 0; i < 4;  ++i) a4[32 + i] = e4[i];
    }
    __syncthreads();

    // ---- WMMA GEMM: 32x144 @ 144x96, software-pipelined --------------------
    const float* Arow0 = Abuf + (size_t)wave * 32 * AM_STR + ln16 * AM_STR;
    const float* Arow1 = Arow0 + 16 * AM_STR;
    const float* Brow  = Wt + ln16 * AM_STR;
    const int    koff  = hi * 2;   // lanes 0-15: K,K+1 ; lanes 16-31: K+2,K+3

    v8f z = {};
    v8f c00=z,c01=z,c02=z,c03=z,c04=z,c05=z;
    v8f c10=z,c11=z,c12=z,c13=z,c14=z,c15=z;

    v2f a0 = *(const v2f*)(Arow0 + koff);
    v2f a1 = *(const v2f*)(Arow1 + koff);
    v2f b0 = *(const v2f*)(Brow + 0 * 16 * AM_STR + koff);
    v2f b1 = *(const v2f*)(Brow + 1 * 16 * AM_STR + koff);
    v2f b2 = *(const v2f*)(Brow + 2 * 16 * AM_STR + koff);
    v2f b3 = *(const v2f*)(Brow + 3 * 16 * AM_STR + koff);
    v2f b4 = *(const v2f*)(Brow + 4 * 16 * AM_STR + koff);
    v2f b5 = *(const v2f*)(Brow + 5 * 16 * AM_STR + koff);

    for (int kb = 0; kb < KM; kb += 4) {
        const int kn = (kb + 4 < KM ? kb + 4 : 0) + koff;
        v2f na0 = *(const v2f*)(Arow0 + kn);
        v2f na1 = *(const v2f*)(Arow1 + kn);
        v2f nb0 = *(const v2f*)(Brow + 0 * 16 * AM_STR + kn);
        v2f nb1 = *(const v2f*)(Brow + 1 * 16 * AM_STR + kn);
        v2f nb2 = *(const v2f*)(Brow + 2 * 16 * AM_STR + kn);
        v2f nb3 = *(const v2f*)(Brow + 3 * 16 * AM_STR + kn);
        v2f nb4 = *(const v2f*)(Brow + 4 * 16 * AM_STR + kn);
        v2f nb5 = *(const v2f*)(Brow + 5 * 16 * AM_STR + kn);
        c00 = WMMA_F32(a0, b0, c00);  c01 = WMMA_F32(a0, b1, c01);
        c02 = WMMA_F32(a0, b2, c02);  c03 = WMMA_F32(a0, b3, c03);
        c04 = WMMA_F32(a0, b4, c04);  c05 = WMMA_F32(a0, b5, c05);
        c10 = WMMA_F32(a1, b0, c10);  c11 = WMMA_F32(a1, b1, c11);
        c12 = WMMA_F32(a1, b2, c12);  c13 = WMMA_F32(a1, b3, c13);
        c14 = WMMA_F32(a1, b4, c14);  c15 = WMMA_F32(a1, b5, c15);
        a0 = na0; a1 = na1;
        b0 = nb0; b1 = nb1; b2 = nb2; b3 = nb3; b4 = nb4; b5 = nb5;
    }

    // ---- bias + relu, stage 32x96 back into this wave's A region -----------
    {
        float* srow = Abuf + (size_t)wave * 32 * AM_STR;
        v8f alo[6] = {c00,c01,c02,c03,c04,c05};
        v8f ahi[6] = {c10,c11,c12,c13,c14,c15};
        #pragma unroll
        for (int nt = 0; nt < 6; ++nt) {
            int n = nt * 16 + ln16;
            float bias = bm[n];
            #pragma unroll
            for (int r = 0; r < 8; ++r) {
                int m = r + 8 * hi;                 // C layout: M = r + 8*(lane>=16)
                float v0 = alo[nt][r] + bias;
                float v1 = ahi[nt][r] + bias;
                srow[m * AM_STR + n]        = v0 > 0.0f ? v0 : 0.0f;
                srow[(m + 16) * AM_STR + n] = v1 > 0.0f ? v1 : 0.0f;
            }
        }
    }
    __syncthreads();

    // ---- LayerNorm stats: one row per lane (32 rows / 32 lanes) ------------
    {
        const float* row = Abuf + (size_t)wave * 32 * AM_STR + lane * AM_STR;
        float s = 0.0f, s2 = 0.0f;
        for (int n = 0; n < FIL; ++n) { float v = row[n]; s += v; s2 += v * v; }
        float mu  = s * (1.0f / FIL);
        float var = s2 * (1.0f / FIL) - mu * mu;
        stats[(wave * 32 + lane) * 2 + 0] = mu;
        stats[(wave * 32 + lane) * 2 + 1] = rsqrtf(var + LN_EPS);
    }
    __syncthreads();

    // ---- normalize, store messages, scatter-add (coalesced per row) --------
    {
        float g0 = gamma[lane], g1 = gamma[lane + 32], g2 = gamma[lane + 64];
        float p0 = beta[lane],  p1 = beta[lane + 32],  p2 = beta[lane + 64];
        for (int m = 0; m < 32; ++m) {
            float mu   = stats[(wave * 32 + m) * 2 + 0];
            float rstd = stats[(wave * 32 + m) * 2 + 1];
            int   dst  = eidx[(wave * 32 + m) * 2 + 1];
            const float* row = Abuf + (size_t)wave * 32 * AM_STR + m * AM_STR;
            float* mout = msg_out + ((size_t)b * EE + (e0 + m)) * FIL;
            float* arow = agg     + ((size_t)b * NN + dst) * FIL;
            float v0 = (row[lane]      - mu) * rstd * g0 + p0;
            float v1 = (row[lane + 32] - mu) * rstd * g1 + p1;
            float v2 = (row[lane + 64] - mu) * rstd * g2 + p2;
            mout[lane]      = v0;  atomicAdd(&arow[lane],      v0);
            mout[lane + 32] = v1;  atomicAdd(&arow[lane + 32], v1);
            mout[lane + 64] = v2;  atomicAdd(&arow[lane + 64], v2);
        }
    }
}

// ---------------------------------------------------------------------------
// Update kernel: 8 waves/block, one 32-node tile per wave. grid = 157.
// ---------------------------------------------------------------------------
__global__ void __launch_bounds__(256) upd_kernel(
    const float* __restrict__ nodes,      // [B][N][64]
    const float* __restrict__ agg,        // [B][N][96]
    const float* __restrict__ Wu,         // [160][96] row-major
    const float* __restrict__ bu,         // [96]
    const float* __restrict__ gamma,      // [96]
    const float* __restrict__ beta,       // [96]
    float*       __restrict__ upd_out)    // [B][N][96]
{
    extern __shared__ float smem[];
    float* Wt    = smem;                              // 96 * 164 floats
    float* Abuf  = Wt + FIL * AU_STR;                 // 256 * 164 floats
    float* stats = Abuf + 256 * AU_STR;               // 256 * 2 floats

    const int tid  = threadIdx.x;
    const int wave = tid >> 5;
    const int lane = tid & 31;
    const int ln16 = lane & 15;
    const int hi   = lane >> 4;

    const int t  = blockIdx.x * 8 + wave;
    const int b  = t / NTP32;
    const int n0 = (t % NTP32) * 32;

    // ---- stage W^T ---------------------------------------------------------
    for (int i = tid; i < KU * FIL; i += 256) {
        int k = i / FIL, n = i % FIL;
        Wt[n * AU_STR + k] = Wu[i];
    }

    // ---- gather A rows: [nodes[n] | aggregated[n]] (zeros past N) ----------
    {
        int r  = tid;                                 // 0..255
        int tw = blockIdx.x * 8 + (r >> 5);
        int bb = tw / NTP32;
        int nn = (tw % NTP32) * 32 + (r & 31);
        bool valid = nn < NN;
        int ns = valid ? nn : 0;
        const float4* s4 = (const float4*)(nodes + ((size_t)bb * NN + ns) * FF);
        const float4* g4 = (const float4*)(agg   + ((size_t)bb * NN + ns) * FIL);
        float4* a4 = (float4*)(Abuf + (size_t)(r >> 5) * 32 * AU_STR + (r & 31) * AU_STR);
        float4 zf = make_float4(0.f, 0.f, 0.f, 0.f);
        #pragma unroll
        for (int i = 0; i < 16; ++i) a4[i]      = valid ? s4[i] : zf;
        #pragma unroll
        for (int i = 0; i < 24; ++i) a4[16 + i] = valid ? g4[i] : zf;
    }
    __syncthreads();

    // ---- WMMA GEMM: 32x160 @ 160x96, software-pipelined --------------------
    const float* Arow0 = Abuf + (size_t)wave * 32 * AU_STR + ln16 * AU_STR;
    const float* Arow1 = Arow0 + 16 * AU_STR;
    const float* Brow  = Wt + ln16 * AU_STR;
    const int    koff  = hi * 2;

    v8f z = {};
    v8f c00=z,c01=z,c02=z,c03=z,c04=z,c05=z;
    v8f c10=z,c11=z,c12=z,c13=z,c14=z,c15=z;

    v2f a0 = *(const v2f*)(Arow0 + koff);
    v2f a1 = *(const v2f*)(Arow1 + koff);
    v2f b0 = *(const v2f*)(Brow + 0 * 16 * AU_STR + koff);
    v2f b1 = *(const v2f*)(Brow + 1 * 16 * AU_STR + koff);
    v2f b2 = *(const v2f*)(Brow + 2 * 16 * AU_STR + koff);
    v2f b3 = *(const v2f*)(Brow + 3 * 16 * AU_STR + koff);
    v2f b4 = *(const v2f*)(Brow + 4 * 16 * AU_STR + koff);
    v2f b5 = *(const v2f*)(Brow + 5 * 16 * AU_STR + koff);

    for (int kb = 0; kb < KU; kb += 4) {
        const int kn = (kb + 4 < KU ? kb + 4 : 0) + koff;
        v2f na0 = *(const v2f*)(Arow0 + kn);
        v2f na1 = *(const v2f*)(Arow1 + kn);
        v2f nb0 = *(const v2f*)(Brow + 0 * 16 * AU_STR + kn);
        v2f nb1 = *(const v2f*)(Brow + 1 * 16 * AU_STR + kn);
        v2f nb2 = *(const v2f*)(Brow + 2 * 16 * AU_STR + kn);
        v2f nb3 = *(const v2f*)(Brow + 3 * 16 * AU_STR + kn);
        v2f nb4 = *(const v2f*)(Brow + 4 * 16 * AU_STR + kn);
        v2f nb5 = *(const v2f*)(Brow + 5 * 16 * AU_STR + kn);
        c00 = WMMA_F32(a0, b0, c00);  c01 = WMMA_F32(a0, b1, c01);
        c02 = WMMA_F32(a0, b2, c02);  c03 = WMMA_F32(a0, b3, c03);
        c04 = WMMA_F32(a0, b4, c04);  c05 = WMMA_F32(a0, b5, c05);
        c10 = WMMA_F32(a1, b0, c10);  c11 = WMMA_F32(a1, b1, c11);
        c12 = WMMA_F32(a1, b2, c12);  c13 = WMMA_F32(a1, b3, c13);
        c14 = WMMA_F32(a1, b4, c14);  c15 = WMMA_F32(a1, b5, c15);
        a0 = na0; a1 = na1;
        b0 = nb0; b1 = nb1; b2 = nb2; b3 = nb3; b4 = nb4; b5 = nb5;
    }

    // ---- bias + relu, stage 32x96 back into LDS ----------------------------
    {
        float* srow = Abuf + (size_t)wave * 32 * AU_STR;
        v8f alo[6] = {c00,c01,c02,c03,c04,c05};
        v8f ahi[6] = {c10,c11,c12,c13,c14,c15};
        #pragma unroll
        for (int nt = 0; nt < 6; ++nt) {
            int n = nt * 16 + ln16;
            float bias = bu[n];
            #pragma unroll
            for (int r = 0; r < 8; ++r) {
                int m = r + 8 * hi;
                float v0 = alo[nt][r] + bias;
                float v1 = ahi[nt][r] + bias;
                srow[m * AU_STR + n]        = v0 > 0.0f ? v0 : 0.0f;
                srow[(m + 16) * AU_STR + n] = v1 > 0.0f ? v1 : 0.0f;
            }
        }
    }
    __syncthreads();

    // ---- LayerNorm stats ---------------------------------------------------
    {
        const float* row = Abuf + (size_t)wave * 32 * AU_STR + lane * AU_STR;
        float s = 0.0f, s2 = 0.0f;
        for (int n = 0; n < FIL; ++n) { float v = row[n]; s += v; s2 += v * v; }
        float mu  = s * (1.0f / FIL);
        float var = s2 * (1.0f / FIL) - mu * mu;
        stats[(wave * 32 + lane) * 2 + 0] = mu;
        stats[(wave * 32 + lane) * 2 + 1] = rsqrtf(var + LN_EPS);
    }
    __syncthreads();

    // ---- normalize + store updated nodes (coalesced per row) ---------------
    {
        float g0 = gamma[lane], g1 = gamma[lane + 32], g2 = gamma[lane + 64];
        float p0 = beta[lane],  p1 = beta[lane + 32],  p2 = beta[lane + 64];
        for (int m = 0; m < 32; ++m) {
            if (n0 + m >= NN) break;                  // uniform per wave
            float mu   = stats[(wave * 32 + m) * 2 + 0];
            float rstd = stats[(wave * 32 + m) * 2 + 1];
            const float* row = Abuf + (size_t)wave * 32 * AU_STR + m * AU_STR;
            float* uout = upd_out + ((size_t)b * NN + (n0 + m)) * FIL;
            uout[lane]      = (row[lane]      - mu) * rstd * g0 + p0;
            uout[lane + 32] = (row[lane + 32] - mu) * rstd * g1 + p1;
            uout[lane + 64] = (row[lane + 64] - mu) * rstd * g2 + p2;
        }
    }
}

extern "C" void kernel_launch(void* const* d_in, const int* in_sizes, int n_in,
                              void* d_out, int out_size, void* d_ws, size_t ws_size,
                              hipStream_t stream) {
    const float* nodes = (const float*)d_in[0];
    const float* efeat = (const float*)d_in[1];
    const int*   edges = (const int*)d_in[2];
    const float* Wm    = (const float*)d_in[3];
    const float* bm    = (const float*)d_in[4];
    const float* g_m   = (const float*)d_in[5];
    const float* b_m   = (const float*)d_in[6];
    const float* Wu    = (const float*)d_in[7];
    const float* bu    = (const float*)d_in[8];
    const float* g_u   = (const float*)d_in[9];
    const float* b_u   = (const float*)d_in[10];

    float* out_upd = (float*)d_out;                              // [B][N][96]
    float* out_msg = out_upd + (size_t)BB * NN * FIL;            // [B][E][96]
    float* agg     = (float*)d_ws;                               // [B][N][96]

    // 1) zero the aggregation scratch
    const int agg_n = BB * NN * FIL;                             // 3,840,000
    zero_f32<<<(agg_n + 255) / 256, 256, 0, stream>>>(agg, agg_n);

    // 2) messages: gather + GEMM(144x96) + relu + LN + scatter-add
    const size_t msg_lds = (size_t)(FIL * AM_STR + 256 * AM_STR) * 4
                         + 512 * 4 + 256 * 2 * 4;                // ~212 KB
    msg_kernel<<<(BB * ETP32) / 8, 256, msg_lds, stream>>>(
        nodes, efeat, edges, Wm, bm, g_m, b_m, out_msg, agg);

    // 3) update: concat + GEMM(160x96) + relu + LN
    const size_t upd_lds = (size_t)(FIL * AU_STR + 256 * AU_STR) * 4
                         + 256 * 2 * 4;                          // ~233 KB
    upd_kernel<<<(BB * NTP32) / 8, 256, upd_lds, stream>>>(
        nodes, agg, Wu, bu, g_u, b_u, out_upd);
}